// PhonemeCTCdecoder_74766790689112
// MI455X (gfx1250) — compile-verified
//
#include <hip/hip_runtime.h>
#include <hip/hip_bf16.h>
#include <stdint.h>

#define LOGZERO_F (-4290774016.0f)

typedef unsigned int u32;
typedef unsigned long long u64;
typedef __attribute__((ext_vector_type(4))) u32 u32x4;
typedef __attribute__((ext_vector_type(8))) int i32x8;
typedef __attribute__((ext_vector_type(4))) int i32x4;
typedef __attribute__((ext_vector_type(2))) float f32x2;
typedef __attribute__((ext_vector_type(8))) float f32x8;

__device__ __forceinline__ float lae(float a, float b) {
    // logaddexp, matches jnp.logaddexp numerics (underflow -> exact max)
    float mx = fmaxf(a, b);
    float mn = fminf(a, b);
    return mx + log1pf(__expf(mn - mx));
}

// ---------------------------------------------------------------------------
// Tensor Data Mover: build a D# (CDNA5 ISA) and issue
// TENSOR_LOAD_TO_LDS. data_size fixed to 4 bytes (f32). Up to 3D tiles.
//   dims/strides in elements; lds_addr in bytes.
// 6-arg builtin form (clang-23 / therock-10.0 headers).
// ---------------------------------------------------------------------------
__device__ __forceinline__ void tdm_load_to_lds(
    u32 lds_addr, const void* gptr,
    u32 td0, u32 td1, u32 td2,          // tensor_dim0..2
    u64 st0, u64 st1,                   // tensor_dim0_stride, tensor_dim1_stride
    u32 tl0, u32 tl1, u32 tl2)          // tile_dim0..2
{
    u64 ga = (u64)gptr;
    u32x4 g0;
    g0.x = 1u;                                         // count=1, user mode
    g0.y = lds_addr;                                   // LDS byte address
    g0.z = (u32)ga;                                    // global_addr[31:0]
    g0.w = (u32)((ga >> 32) & 0x01FFFFFFull) | (2u << 30); // addr[56:32] | type=2
    i32x8 g1;
    g1[0] = (int)(2u << 16);                           // data_size = 4 bytes
    g1[1] = (int)((td0 & 0xFFFFu) << 16);              // tensor_dim0[15:0]
    g1[2] = (int)(((td0 >> 16) & 0xFFFFu) | ((td1 & 0xFFFFu) << 16));
    g1[3] = (int)(((td1 >> 16) & 0xFFFFu) | ((tl0 & 0xFFFFu) << 16)); // tile_dim0
    g1[4] = (int)((tl1 & 0xFFFFu) | ((tl2 & 0xFFFFu) << 16));         // tile_dim1/2
    g1[5] = (int)(u32)st0;                             // dim0_stride[31:0]
    g1[6] = (int)(((u32)(st0 >> 32) & 0xFFFFu) | (((u32)st1 & 0xFFFFu) << 16));
    g1[7] = (int)(u32)(st1 >> 16);                     // dim1_stride[47:16]
    i32x4 g2 = { (int)td2, 0, 0, 0 };                  // tensor_dim2; tile_dim3=0
    i32x4 g3 = { 0, 0, 0, 0 };
    i32x8 gx = { 0, 0, 0, 0, 0, 0, 0, 0 };             // extra operand: zero-filled
    __builtin_amdgcn_tensor_load_to_lds(g0, g1, g2, g3, gx, 0);
}

// ---------------------------------------------------------------------------
// Kernel A: per-(b,v) sequential scan over t + online logsumexp over valid t.
// grid = B blocks, block = V (=128) threads (4 wave32 waves).
// Strided per-b slices of lastP and x[...,blank] are fetched by the TDM.
// ---------------------------------------------------------------------------
__global__ __launch_bounds__(128) void ctc_scan_kernel(
    const float* __restrict__ x,            // (B,T,V)
    const float* __restrict__ lastP,        // (2,T,B)
    const float* __restrict__ prefP_scalar, // (B)
    const int*   __restrict__ xl,           // (B)
    const int*   __restrict__ last_pred,    // (B)
    const int*   __restrict__ cur_len_p,    // scalar
    const int*   __restrict__ blank_p,      // scalar
    float* __restrict__ sOut,               // (B,V): curP - prefP_scalar
    float* __restrict__ eosOut,             // (B)
    int B, int T, int V)
{
    const int b = blockIdx.x;
    const int v = threadIdx.x;
    const int blank = *blank_p;
    int start = *cur_len_p - 1;
    if (start > T - 1) start = T - 1;
    if (start < 0) start = 0;

    __shared__ float s_lp[2 * 512];  // [plane][t] = lastP[plane][t][b]
    __shared__ float s_lsum[512];    // logaddexp(lastP[0], lastP[1])
    __shared__ float s_xb[512];      // x[b][t][blank]

    if (threadIdx.x < 32) {  // wave 0 issues the DMA descriptors
        // lastP viewed as 3D tensor (dim0=B, dim1=T, dim2=2); tile 1 x T x 2 at column b
        tdm_load_to_lds((u32)(uintptr_t)s_lp, lastP + b,
                        (u32)B, (u32)T, 2u,
                        (u64)B, (u64)T * (u64)B,
                        1u, (u32)T, 2u);
        // x[b] viewed as 2D tensor (dim0=V, dim1=T); tile 1 x T at column `blank`
        tdm_load_to_lds((u32)(uintptr_t)s_xb, x + (size_t)b * T * V + blank,
                        (u32)V, (u32)T, 0u,
                        (u64)V, 0ull,
                        1u, (u32)T, 0u);
        __builtin_amdgcn_s_wait_tensorcnt(0);
    }
    __syncthreads();

    for (int t = threadIdx.x; t < T; t += blockDim.x)
        s_lsum[t] = lae(s_lp[t], s_lp[T + t]);
    __syncthreads();

    const int myxl = xl[b];
    const int lp   = last_pred[b];
    const float* xrow = x + (size_t)b * T * V + v;   // stride V in t

    float Pn, Pb;
    float m = LOGZERO_F, ssum = 0.0f;   // online logsumexp state
    int lo;
    if (start == 0) {
        Pn = xrow[0];          // Pn0 = Xn[0]
        Pb = LOGZERO_F;        // Pb0
        lo = 1;
        if (0 < myxl) {
            float c = lae(Pn, Pb);   // cur_all[0]
            if (c > m) { ssum = ssum * __expf(m - c) + 1.0f; m = c; }
            else       { ssum += __expf(c - m); }
        }
    } else {
        Pn = LOGZERO_F;
        Pb = LOGZERO_F;
        lo = start;
    }

    float xn = (lo < T) ? xrow[(size_t)lo * V] : 0.0f;
    for (int t = lo; t < T; ++t) {
        // software pipelining + prefetch ahead of the latency-bound recurrence
        float xn_next = (t + 1 < T) ? xrow[(size_t)(t + 1) * V] : 0.0f;
        if (t + 8 < T) __builtin_prefetch(xrow + (size_t)(t + 8) * V, 0, 0);

        float pref = (lp == v) ? s_lp[T + (t - 1)] : s_lsum[t - 1];
        float Pn2 = lae(Pn, pref) + xn;
        float Pb2 = lae(Pn, Pb) + s_xb[t];
        Pn = Pn2; Pb = Pb2;
        if (t < myxl) {
            float c = lae(Pn2, Pb2);
            if (c > m) { ssum = ssum * __expf(m - c) + 1.0f; m = c; }
            else       { ssum += __expf(c - m); }
        }
        xn = xn_next;
    }

    float curP = (ssum > 0.0f) ? (m + __logf(ssum)) : LOGZERO_F;
    sOut[(size_t)b * V + v] = curP - prefP_scalar[b];

    if (v == 0) {
        // eosP[b] = lastPsum[xl[b]-1, b]
        float e = 0.0f;
        int te = myxl - 1;
        if (te >= 0 && te < T) e = s_lsum[te];
        eosOut[b] = e;
    }
}

// ---------------------------------------------------------------------------
// Kernel B: finalP[b,w] = max over active v of s[b,v] — tropical (max,+)
// product. The sparse structure (per-column 128-bit masks) is computed with
// f32 WMMA: h[m][w] = sum_j 2^j * p2w[16m+j, w] is an exact integer < 2^16,
// so V_WMMA_F32_16X16X4_F32 accumulation reproduces the bit pattern exactly.
// The tropical max itself runs on VALU over the set bits only (~2% density).
// ---------------------------------------------------------------------------
#define WT 128
#define BT 16

__global__ __launch_bounds__(256) void ctc_maskmax_kernel(
    const float* __restrict__ p2w,    // (V,W), values exactly {0.0f, 1.0f}
    const float* __restrict__ sIn,    // (B,V)
    const float* __restrict__ eosIn,  // (B)
    const int*   __restrict__ blank_p,
    const int*   __restrict__ eos_p,
    float* __restrict__ out,          // (B,W)
    int B, int V, int W)
{
    __shared__ float smMask[128 * WT];  // 64 KB float tile (V rows x WT cols)
    __shared__ uint4 smPacked[WT];      // 2 KB: per w-column 128-bit active-v mask
    __shared__ float smS[BT * 128];     // 8 KB (BT rows x V cols)

    const int w0  = blockIdx.x * WT;
    const int b0  = blockIdx.y * BT;
    const int tid = threadIdx.x;

    if (tid < 32) {  // wave 0: one TDM descriptor moves the whole 64KB tile
        // p2w viewed as 2D tensor (dim0=W, dim1=V); tile WT x V at column w0.
        // OOB columns (w0+wi >= W) read as 0 -> mask bit 0 -> LOGZERO, exact.
        tdm_load_to_lds((u32)(uintptr_t)smMask, p2w + w0,
                        (u32)W, (u32)V, 0u,
                        (u64)W, 0ull,
                        (u32)WT, (u32)V, 0u);
    }
    // s tile: per-lane async global->LDS B128 chunks (ASYNCcnt path)
    for (int i = tid; i < BT * (V / 4); i += 256) {
        int bi = i / (V / 4);
        int vi = (i % (V / 4)) * 4;
        if (b0 + bi < B) {
            const float* g = sIn + (size_t)(b0 + bi) * V + vi;
            unsigned lds = (unsigned)(uintptr_t)(&smS[bi * 128 + vi]);
            asm volatile("global_load_async_to_lds_b128 %0, %1, off"
                         :: "v"(lds), "v"(g) : "memory");
        }
    }
    asm volatile("s_wait_asynccnt 0x0" ::: "memory");
    if (tid < 32) __builtin_amdgcn_s_wait_tensorcnt(0);
    __syncthreads();

    // ---- Bit-pack via WMMA: D(16x16) = W16(16xK=128) x p2w_tile(128 x 16w).
    // Wave `wave` owns w-columns [wave*16, wave*16+16). A-matrix rows m hold
    // weights 2^(v&15) on v in [16m,16m+16) (zero for v>=V); rows 8..15 and
    // all padding contribute exact zeros. All partial sums are integers <2^16
    // -> f32 accumulation is bit-exact.
    {
        const int wave = tid >> 5;          // 0..7 -> 16-wide w chunk
        const int lane = tid & 31;
        const int w0c  = wave * 16;
        const int mn   = lane & 15;         // A row M / B,D column N
        const int kh   = lane >> 4;         // 0: K=0,1 ; 1: K=2,3 (16x4 layout)
        f32x8 acc = {0.f,0.f,0.f,0.f,0.f,0.f,0.f,0.f};
        for (int kc = 0; kc < 32; ++kc) {   // K = 128 in steps of 4
            int v0 = kc * 4 + kh * 2;
            int v1 = v0 + 1;
            f32x2 a, bm;
            a.x = (v0 < V && (v0 >> 4) == mn) ? (float)(1 << (v0 & 15)) : 0.0f;
            a.y = (v1 < V && (v1 >> 4) == mn) ? (float)(1 << (v1 & 15)) : 0.0f;
            bm.x = smMask[v0 * WT + w0c + mn];
            bm.y = smMask[v1 * WT + w0c + mn];
            acc = __builtin_amdgcn_wmma_f32_16x16x4_f32(
                      false, a, false, bm, (short)0, acc, false, false);
        }
        if (lane < 16) {    // lane n holds D[m][n], m=0..7 -> all 8 halves
            unsigned h0 = (unsigned)acc[0], h1 = (unsigned)acc[1];
            unsigned h2 = (unsigned)acc[2], h3 = (unsigned)acc[3];
            unsigned h4 = (unsigned)acc[4], h5 = (unsigned)acc[5];
            unsigned h6 = (unsigned)acc[6], h7 = (unsigned)acc[7];
            uint4 pw;
            pw.x = h0 | (h1 << 16);
            pw.y = h2 | (h3 << 16);
            pw.z = h4 | (h5 << 16);
            pw.w = h6 | (h7 << 16);
            smPacked[w0c + lane] = pw;
        }
    }
    __syncthreads();

    const int w    = tid & (WT - 1);       // w column within tile
    const int bi0  = (tid >> 7) * (BT / 2);// 0 or 8: which half of the b-rows
    const int gw   = w0 + w;
    const int eosI = *eos_p;
    const int blkI = *blank_p;

    float best[BT / 2];
    #pragma unroll
    for (int i = 0; i < BT / 2; ++i) best[i] = LOGZERO_F;

    uint4 pm = smPacked[w];                // one ds_load_b128
    unsigned words[4] = { pm.x, pm.y, pm.z, pm.w };
    #pragma unroll
    for (int k = 0; k < 4; ++k) {
        unsigned mword = words[k];
        while (mword) {                    // ~2.6 iterations expected at 2% density
            int j = __builtin_ctz(mword);
            mword &= mword - 1;
            int v = k * 32 + j;
            #pragma unroll
            for (int i = 0; i < BT / 2; ++i)
                best[i] = fmaxf(best[i], smS[(bi0 + i) * 128 + v]);
        }
    }

    #pragma unroll
    for (int i = 0; i < BT / 2; ++i) {
        int b = b0 + bi0 + i;
        if (b >= B || gw >= W) continue;
        float r = best[i];
        if (gw == blkI || gw == 1) r = LOGZERO_F;
        if (gw == eosI)            r = eosIn[b];
        out[(size_t)b * W + gw] = r;       // lane-consecutive -> coalesced
    }
}

extern "C" void kernel_launch(void* const* d_in, const int* in_sizes, int n_in,
                              void* d_out, int out_size, void* d_ws, size_t ws_size,
                              hipStream_t stream) {
    const float* x         = (const float*)d_in[0];
    const float* lastP     = (const float*)d_in[1];
    const float* prefP     = (const float*)d_in[2];
    const float* p2w       = (const float*)d_in[3];
    const int*   xl        = (const int*)d_in[4];
    const int*   last_pred = (const int*)d_in[5];
    const int*   cur_len   = (const int*)d_in[6];
    const int*   blank     = (const int*)d_in[7];
    const int*   eos       = (const int*)d_in[8];

    const int B = in_sizes[4];                 // xl: (B,)
    const int T = in_sizes[1] / (2 * B);       // lastP: (2,T,B)
    const int V = in_sizes[0] / (B * T);       // x: (B,T,V)
    const int W = in_sizes[3] / V;             // p2w: (V,W)

    float* sBuf   = (float*)d_ws;              // (B,V)
    float* eosBuf = sBuf + (size_t)B * V;      // (B)

    ctc_scan_kernel<<<B, V, 0, stream>>>(x, lastP, prefP, xl, last_pred,
                                         cur_len, blank, sBuf, eosBuf, B, T, V);

    dim3 grid((W + WT - 1) / WT, (B + BT - 1) / BT);
    ctc_maskmax_kernel<<<grid, 256, 0, stream>>>(p2w, sBuf, eosBuf, blank, eos,
                                                 (float*)d_out, B, V, W);
}